// BayesianLinear_57234734186710
// MI455X (gfx1250) — compile-verified
//
#include <hip/hip_runtime.h>
#include <cmath>

#define BATCH 1024
#define IN_F  512
#define OUT_F 512

typedef float v2f __attribute__((ext_vector_type(2)));
typedef float v8f __attribute__((ext_vector_type(8)));

// ---------------------------------------------------------------------------
// k0: sigma[o,i] = exp(0.5 * weight_logvar[o,i])   (1 MB, stays in L2)
// ---------------------------------------------------------------------------
__global__ void std_kernel(const float* __restrict__ wl, float* __restrict__ stdw) {
    int idx = blockIdx.x * blockDim.x + threadIdx.x;
    stdw[idx] = expf(0.5f * wl[idx]);
}

// ---------------------------------------------------------------------------
// k1: out[b,o] = (x @ mu^T)[b,o] + bias_mu[o] + exp(0.5*bias_logvar[o])*eps_b[b,o]
// One wave per 16x16 (batch x out) tile, K looped in steps of 4 via
// V_WMMA_F32_16X16X4_F32. Bias terms are folded into the initial accumulator.
// ---------------------------------------------------------------------------
__global__ void wmma_mu_gemm(const float* __restrict__ x,
                             const float* __restrict__ mu,
                             const float* __restrict__ eps_b,
                             const float* __restrict__ bias_mu,
                             const float* __restrict__ bias_logvar,
                             float* __restrict__ out) {
    const int lane = threadIdx.x;            // 0..31, full wave, EXEC all ones
    const int m0   = blockIdx.x * 16;        // batch tile origin
    const int o0   = blockIdx.y * 16;        // output tile origin
    const int row  = lane & 15;
    const int hi   = lane >> 4;              // 0: lanes 0-15, 1: lanes 16-31
    const int klo  = hi * 2;                 // A/B K origin for this half-wave
    const int col  = row;                    // C/D: N = lane & 15
    const int mbase = hi * 8;                // C/D: M = r + 8*hi

    // Initial C = bias_mu[o] + exp(0.5*bias_logvar[o]) * eps_b[m,o]
    v8f c;
    const int o = o0 + col;
    const float bstd = expf(0.5f * bias_logvar[o]);
    const float bmu  = bias_mu[o];
#pragma unroll
    for (int r = 0; r < 8; ++r) {
        const int m = m0 + mbase + r;
        c[r] = bmu + bstd * eps_b[m * OUT_F + o];
    }

    // A: lanes 0-15 hold K=klo(+0/+1) of row M=lane&15 ; B mirrors with N.
    const float* xrow  = x  + (m0 + row) * IN_F + klo;
    const float* murow = mu + (o0 + row) * IN_F + klo;

#pragma unroll 4
    for (int k = 0; k < IN_F; k += 4) {
        v2f a = *(const v2f*)(xrow + k);     // x[m0+row][k+klo .. +1]
        v2f b = *(const v2f*)(murow + k);    // mu[o0+row][k+klo .. +1]
        c = __builtin_amdgcn_wmma_f32_16x16x4_f32(
                false, a, false, b, (short)0, c, false, false);
    }

#pragma unroll
    for (int r = 0; r < 8; ++r)
        out[(m0 + mbase + r) * OUT_F + o0 + col] = c[r];
}

// ---------------------------------------------------------------------------
// k2: out[b,o] += sum_i sigma[o,i] * eps_w[b,o,i] * x[b,i]
// One block per batch b (256 threads = 8 waves). x[b,:] staged in LDS.
// Wave w streams contiguous rows o = w*64 .. w*64+63; lanes issue coalesced
// float4 loads (512B per wave instruction) of the 1 GiB eps_w stream.
// ---------------------------------------------------------------------------
template <bool PRECOMP>
__global__ void eps_stream_kernel(const float* __restrict__ x,
                                  const float* __restrict__ eps_w,
                                  const float* __restrict__ src,   // sigma or logvar
                                  float* __restrict__ out) {
    __shared__ __align__(16) float xs[IN_F];
    const int b    = blockIdx.x;
    const int tid  = threadIdx.x;
    const int lane = tid & 31;
    const int wave = tid >> 5;

    for (int i = tid; i < IN_F; i += 256) xs[i] = x[b * IN_F + i];
    __syncthreads();

    const float4* xs4      = (const float4*)xs;
    const size_t  eps_base = (size_t)b * OUT_F * IN_F;

    for (int oi = 0; oi < 64; ++oi) {
        const int o = wave * 64 + oi;
        const float4* ep = (const float4*)(eps_w + eps_base + (size_t)o * IN_F);
        const float4* sp = (const float4*)(src + (size_t)o * IN_F);

        float4 acc = make_float4(0.f, 0.f, 0.f, 0.f);
#pragma unroll
        for (int j = 0; j < 4; ++j) {
            const int v = j * 32 + lane;     // element index (j*128 + lane*4)/4
            float4 e  = ep[v];
            float4 s  = sp[v];
            if (!PRECOMP) {
                s.x = expf(0.5f * s.x); s.y = expf(0.5f * s.y);
                s.z = expf(0.5f * s.z); s.w = expf(0.5f * s.w);
            }
            float4 xv = xs4[v];
            acc.x = fmaf(s.x * e.x, xv.x, acc.x);
            acc.y = fmaf(s.y * e.y, xv.y, acc.y);
            acc.z = fmaf(s.z * e.z, xv.z, acc.z);
            acc.w = fmaf(s.w * e.w, xv.w, acc.w);
        }
        float dot = (acc.x + acc.y) + (acc.z + acc.w);
#pragma unroll
        for (int m = 16; m > 0; m >>= 1)
            dot += __shfl_xor(dot, m, 32);   // wave32 butterfly reduction
        if (lane == 0)
            out[b * OUT_F + o] += dot;
    }
}

// ---------------------------------------------------------------------------
extern "C" void kernel_launch(void* const* d_in, const int* in_sizes, int n_in,
                              void* d_out, int out_size, void* d_ws, size_t ws_size,
                              hipStream_t stream) {
    const float* x    = (const float*)d_in[0];
    const float* epsw = (const float*)d_in[1];
    const float* epsb = (const float*)d_in[2];
    const float* wmu  = (const float*)d_in[3];
    const float* wlv  = (const float*)d_in[4];
    const float* bmu  = (const float*)d_in[5];
    const float* blv  = (const float*)d_in[6];
    float* out = (float*)d_out;

    const size_t std_bytes = (size_t)OUT_F * IN_F * sizeof(float);
    const bool   precomp   = ws_size >= std_bytes;
    float*       stdw      = (float*)d_ws;

    if (precomp)
        std_kernel<<<(OUT_F * IN_F) / 256, 256, 0, stream>>>(wlv, stdw);

    wmma_mu_gemm<<<dim3(BATCH / 16, OUT_F / 16), 32, 0, stream>>>(
        x, wmu, epsb, bmu, blv, out);

    if (precomp)
        eps_stream_kernel<true><<<BATCH, 256, 0, stream>>>(x, epsw, stdw, out);
    else
        eps_stream_kernel<false><<<BATCH, 256, 0, stream>>>(x, epsw, wlv, out);
}